// Model_37726992728720
// MI455X (gfx1250) — compile-verified
//
#include <hip/hip_runtime.h>
#include <math.h>

// ---- problem constants (match reference) ----
#define NODES       100
#define NGRAPH      1000
#define EPG         1600      // edges per graph
#define NFEAT       32
#define KPOOL       30
#define NCLS        10
#define OUTLEN      16
#define NITER       3

typedef float v2f __attribute__((ext_vector_type(2)));
typedef float v8f __attribute__((ext_vector_type(8)));

// ---- LDS layout (float offsets) ----
#define OFF_EDGE   0          // 1600 u32 packed (ls<<16|ld)
#define OFF_DINV   1600       // 100
#define OFF_XB     1700       // 112x32 (rows >=100 zero)
#define OFF_HB     5284       // 112x32
#define OFF_AC     8868       // 112x32 (also: edge-index async staging at start)
#define OFF_FEAT   12452      // 100 x 104 (cols 0..95 = x1..x3, 96 = x4)
#define OFF_POOL   22852      // 32 x 104  (rows>=30, cols>=97 zero)
#define OFF_PRI    26180      // 32 x 160  (priors, rows>=30 zero)
#define OFF_WB     31300      // 1024 W + 32 bias
#define OFF_SEL    32356      // 30 ints (padded to 32)
#define OFF_PRSQ   32388      // 300
#define OFF_SIM    32688      // 300
#define OFF_PROB   32988      // 300
#define OFF_ROUT   33288      // 160 (out caps)
#define OFF_OSQ    33448      // 10
#define OFF_MAX    33458      // 10
#define OFF_SUM    33468      // 10
#define SMEM_FLOATS 33478
#define SMEM_BYTES  (SMEM_FLOATS * 4)

// ---- CDNA5 async global->LDS copies (ASYNCcnt path), portable inline asm ----
// LDS offset = low 32 bits of the flat shared address (ISA: LDS aperture maps addr[31:0]).
__device__ __forceinline__ unsigned lds_off(const void* p) {
    return (unsigned)(unsigned long long)p;
}
__device__ __forceinline__ void async_ld_b128(unsigned lds, unsigned goff, unsigned long long base) {
    asm volatile("global_load_async_to_lds_b128 %0, %1, %2"
                 :: "v"(lds), "v"(goff), "s"(base) : "memory");
}
__device__ __forceinline__ void async_ld_b32(unsigned lds, unsigned goff, unsigned long long base) {
    asm volatile("global_load_async_to_lds_b32 %0, %1, %2"
                 :: "v"(lds), "v"(goff), "s"(base) : "memory");
}
__device__ __forceinline__ void wait_async0() {
    asm volatile("s_wait_asynccnt 0" ::: "memory");
}

__global__ __launch_bounds__(256)
void gcn_caps_fused(const float* __restrict__ x,
                    const int*   __restrict__ esrc,
                    const int*   __restrict__ edst,
                    const float* __restrict__ W1, const float* __restrict__ b1,
                    const float* __restrict__ W2, const float* __restrict__ b2,
                    const float* __restrict__ W3, const float* __restrict__ b3,
                    const float* __restrict__ W4, const float* __restrict__ b4,
                    const float* __restrict__ Wcap,
                    float* __restrict__ outv)
{
    extern __shared__ __align__(16) float smem[];
    unsigned* EDGE = (unsigned*)(smem + OFF_EDGE);
    float* DINV = smem + OFF_DINV;
    float* XB   = smem + OFF_XB;
    float* HB   = smem + OFF_HB;
    float* AC   = smem + OFF_AC;
    float* FEAT = smem + OFF_FEAT;
    float* POOL = smem + OFF_POOL;
    float* PRI  = smem + OFF_PRI;
    float* WB   = smem + OFF_WB;
    int*   SEL  = (int*)(smem + OFF_SEL);
    float* PRSQ = smem + OFF_PRSQ;
    float* SIM  = smem + OFF_SIM;
    float* PROB = smem + OFF_PROB;
    float* ROUT = smem + OFF_ROUT;
    float* OSQ  = smem + OFF_OSQ;
    float* MAXV = smem + OFF_MAX;
    float* SUMV = smem + OFF_SUM;

    const int tid  = threadIdx.x;
    const int g    = blockIdx.x;
    const int wave = tid >> 5;
    const int lane = tid & 31;
    const int l16  = lane & 15;
    const int khalf = (lane >> 4) * 2;       // lanes 16..31 hold K+2,K+3

    // ---- phase 0: async DMA node features + edge indices into LDS ----
    // x block (100x32 f32 = 800 x b128) -> XB rows 0..99
    {
        const unsigned long long xbase = (unsigned long long)(x + (size_t)g * NODES * NFEAT);
        for (int idx = tid; idx < (NODES * NFEAT) / 4; idx += 256)
            async_ld_b128(lds_off(&XB[idx * 4]), (unsigned)(idx * 16), xbase);
        // edge indices of graph g (stride NGRAPH) -> staged in AC region as ints
        const unsigned long long sbase = (unsigned long long)(esrc + g);
        const unsigned long long dbase = (unsigned long long)(edst + g);
        int* STAGE = (int*)AC;
        for (int j = tid; j < EPG; j += 256) {
            async_ld_b32(lds_off(&STAGE[j]),       (unsigned)(j * NGRAPH * 4), sbase);
            async_ld_b32(lds_off(&STAGE[EPG + j]), (unsigned)(j * NGRAPH * 4), dbase);
        }
    }
    // zero WMMA pad rows + degree accumulator while DMA is in flight
    for (int idx = NODES * NFEAT + tid; idx < 112 * NFEAT; idx += 256) XB[idx] = 0.0f;
    for (int i = tid; i < NODES; i += 256) DINV[i] = 0.0f;
    wait_async0();
    __syncthreads();

    // ---- pack edges to local ids + degree (LDS atomics) ----
    {
        const int* STAGE = (const int*)AC;
        for (int j = tid; j < EPG; j += 256) {
            const int ls = STAGE[j]       - g * NODES;
            const int ld = STAGE[EPG + j] - g * NODES;
            EDGE[j] = ((unsigned)ls << 16) | (unsigned)ld;
            atomicAdd(&DINV[ld], 1.0f);
        }
    }
    __syncthreads();
    for (int i = tid; i < NODES; i += 256) DINV[i] = rsqrtf(DINV[i] + 1.0f);
    __syncthreads();

    // ---- 3 GCN layers 32->32 ----
    const float* Ws[3] = {W1, W2, W3};
    const float* bs[3] = {b1, b2, b3};
    for (int l = 0; l < 3; ++l) {
        // async DMA the 32x32 weight tile (256 x b128); bias via scalar path
        async_ld_b128(lds_off(&WB[tid * 4]), (unsigned)(tid * 16),
                      (unsigned long long)Ws[l]);
        if (tid < 32) WB[1024 + tid] = bs[l][tid];
        for (int idx = tid; idx < 112 * NFEAT; idx += 256) AC[idx] = 0.0f;
        wait_async0();
        __syncthreads();

        // HB = XB @ W  via V_WMMA_F32_16X16X4_F32 (exact f32)
        for (int t = wave; t < 14; t += 8) {            // 7 mtiles x 2 ntiles
            const int mt = t >> 1, nt = t & 1;
            const int rowA = mt * 16 + l16;
            const int colB = nt * 16 + l16;
            v8f c = {};
            #pragma unroll
            for (int k = 0; k < 8; ++k) {
                const int kk = k * 4 + khalf;
                v2f a, b;
                a.x = XB[rowA * 32 + kk];
                a.y = XB[rowA * 32 + kk + 1];
                b.x = WB[kk * 32 + colB];
                b.y = WB[(kk + 1) * 32 + colB];
                c = __builtin_amdgcn_wmma_f32_16x16x4_f32(
                        false, a, false, b, (short)0, c, false, false);
            }
            const int rbase = mt * 16 + ((lane < 16) ? 0 : 8);
            const int colD  = nt * 16 + l16;
            #pragma unroll
            for (int r = 0; r < 8; ++r) HB[(rbase + r) * 32 + colD] = c[r];
        }
        __syncthreads();

        // message passing: AC[dst] += HB[src] * dinv[s]*dinv[d]  (ds_add_f32)
        for (int it = tid; it < EPG * 8; it += 256) {
            const int e  = it >> 3;
            const int fb = (it & 7) * 4;
            const unsigned p = EDGE[e];
            const int ls = p >> 16, ld = p & 0xffff;
            const float w = DINV[ls] * DINV[ld];
            const float4 hv = *(const float4*)&HB[ls * 32 + fb];
            atomicAdd(&AC[ld * 32 + fb + 0], hv.x * w);
            atomicAdd(&AC[ld * 32 + fb + 1], hv.y * w);
            atomicAdd(&AC[ld * 32 + fb + 2], hv.z * w);
            atomicAdd(&AC[ld * 32 + fb + 3], hv.w * w);
        }
        __syncthreads();

        // combine + tanh; feed next layer and FEAT columns
        for (int idx = tid; idx < NODES * 32; idx += 256) {
            const int i = idx >> 5, f = idx & 31;
            const float di = DINV[i];
            const float v = tanhf(AC[idx] + HB[idx] * di * di + WB[1024 + f]);
            XB[idx] = v;
            FEAT[i * 104 + l * 32 + f] = v;
        }
        __syncthreads();
    }

    // ---- layer 4: 32 -> 1 ----
    if (tid < 32) WB[tid] = W4[tid];
    if (tid == 0) WB[32] = b4[0];
    for (int i = tid; i < 128; i += 256) AC[i] = 0.0f;
    __syncthreads();
    for (int i = tid; i < NODES; i += 256) {
        float s = 0.f;
        #pragma unroll
        for (int k = 0; k < 32; ++k) s += XB[i * 32 + k] * WB[k];
        HB[i] = s;
    }
    __syncthreads();
    for (int j = tid; j < EPG; j += 256) {
        const unsigned p = EDGE[j];
        const int ls = p >> 16, ld = p & 0xffff;
        atomicAdd(&AC[ld], HB[ls] * DINV[ls] * DINV[ld]);
    }
    __syncthreads();
    for (int i = tid; i < NODES; i += 256) {
        const float di = DINV[i];
        FEAT[i * 104 + 96] = tanhf(AC[i] + HB[i] * di * di + WB[32]);
    }
    __syncthreads();

    // ---- top-30 pooling by x4 (descending, stable => ties by lower index) ----
    if (tid < NODES) {
        const float v = FEAT[tid * 104 + 96];
        int rank = 0;
        for (int j = 0; j < NODES; ++j) {
            const float vj = FEAT[j * 104 + 96];
            rank += (vj > v) || (vj == v && j < tid);
        }
        if (rank < KPOOL) SEL[rank] = tid;
    }
    __syncthreads();
    for (int idx = tid; idx < 32 * 104; idx += 256) {
        const int r = idx / 104, k = idx - r * 104;
        POOL[idx] = (r < KPOOL && k < 97) ? FEAT[SEL[r] * 104 + k] : 0.0f;
    }
    __syncthreads();

    // ---- priors: POOL(32x104) @ Wcap^T(104x160) via WMMA f32 16x16x4 ----
    for (int t = wave; t < 20; t += 8) {                // 2 mtiles x 10 classes
        const int mt = (t >= 10) ? 1 : 0;
        const int c_t = (t >= 10) ? t - 10 : t;
        const int rowA = mt * 16 + l16;
        const float* wc = Wcap + (c_t * 16 + l16) * 97; // B[k][c*16+e] = Wcap[c][e][k]
        v8f c = {};
        #pragma unroll
        for (int k = 0; k < 26; ++k) {
            const int kk = 4 * k + khalf;
            v2f a, b;
            a.x = POOL[rowA * 104 + kk];
            a.y = POOL[rowA * 104 + kk + 1];
            b.x = (kk     < 97) ? wc[kk]     : 0.0f;
            b.y = (kk + 1 < 97) ? wc[kk + 1] : 0.0f;
            c = __builtin_amdgcn_wmma_f32_16x16x4_f32(
                    false, a, false, b, (short)0, c, false, false);
        }
        const int rbase = mt * 16 + ((lane < 16) ? 0 : 8);
        const int colD  = c_t * 16 + l16;
        #pragma unroll
        for (int r = 0; r < 8; ++r) PRI[(rbase + r) * 160 + colD] = c[r];
    }
    __syncthreads();

    // ---- dynamic routing (3 iterations) ----
    for (int idx = tid; idx < KPOOL * NCLS; idx += 256) {
        const int i = idx / NCLS, cc = idx - i * NCLS;
        const float* p = &PRI[i * 160 + cc * 16];
        float s = 0.f;
        #pragma unroll
        for (int e = 0; e < 16; ++e) s += p[e] * p[e];
        PRSQ[idx] = s;
    }
    for (int idx = tid; idx < NCLS * OUTLEN; idx += 256) {
        float s = 0.f;
        for (int i = 0; i < KPOOL; ++i) s += PRI[i * 160 + idx];
        ROUT[idx] = s * (1.0f / KPOOL);
    }
    __syncthreads();

    for (int iter = 0; iter < NITER; ++iter) {
        if (tid < NCLS) {
            float s = 0.f;
            #pragma unroll
            for (int e = 0; e < 16; ++e) { const float v = ROUT[tid * 16 + e]; s += v * v; }
            OSQ[tid] = s;
        }
        __syncthreads();
        for (int idx = tid; idx < KPOOL * NCLS; idx += 256) {
            const int i = idx / NCLS, cc = idx - i * NCLS;
            const float* p = &PRI[i * 160 + cc * 16];
            const float* o = &ROUT[cc * 16];
            float xy = 0.f;
            #pragma unroll
            for (int e = 0; e < 16; ++e) xy += p[e] * o[e];
            SIM[idx] = xy / (PRSQ[idx] + OSQ[cc] - xy);
        }
        __syncthreads();
        if (tid < NCLS) {
            float m = -1e30f;
            for (int i = 0; i < KPOOL; ++i) m = fmaxf(m, SIM[i * NCLS + tid]);
            float s = 0.f;
            for (int i = 0; i < KPOOL; ++i) s += expf(SIM[i * NCLS + tid] - m);
            MAXV[tid] = m; SUMV[tid] = s;
        }
        __syncthreads();
        for (int idx = tid; idx < KPOOL * NCLS; idx += 256) {
            const int cc = idx % NCLS;
            PROB[idx] = expf(SIM[idx] - MAXV[cc]) / SUMV[cc];
        }
        __syncthreads();
        for (int idx = tid; idx < NCLS * OUTLEN; idx += 256) {
            const int cc = idx >> 4;
            float s = 0.f;
            for (int i = 0; i < KPOOL; ++i) s += PROB[i * NCLS + cc] * PRI[i * 160 + idx];
            ROUT[idx] = s;
        }
        __syncthreads();
    }

    if (tid < NCLS) {
        float s = 0.f;
        #pragma unroll
        for (int e = 0; e < 16; ++e) { const float v = ROUT[tid * 16 + e]; s += v * v; }
        outv[g * NCLS + tid] = sqrtf(s);
    }
}

extern "C" void kernel_launch(void* const* d_in, const int* in_sizes, int n_in,
                              void* d_out, int out_size, void* d_ws, size_t ws_size,
                              hipStream_t stream) {
    const float* x    = (const float*)d_in[0];
    const int*   es   = (const int*)  d_in[1];
    const int*   ed   = (const int*)  d_in[2];
    /* d_in[3] = batch (unused: graphs are contiguous blocks of 100 nodes) */
    const float* W1   = (const float*)d_in[4];
    const float* b1   = (const float*)d_in[5];
    const float* W2   = (const float*)d_in[6];
    const float* b2   = (const float*)d_in[7];
    const float* W3   = (const float*)d_in[8];
    const float* b3   = (const float*)d_in[9];
    const float* W4   = (const float*)d_in[10];
    const float* b4   = (const float*)d_in[11];
    const float* Wcap = (const float*)d_in[12];
    float* outv = (float*)d_out;

    gcn_caps_fused<<<dim3(NGRAPH), dim3(256), SMEM_BYTES, stream>>>(
        x, es, ed, W1, b1, W2, b2, W3, b3, W4, b4, Wcap, outv);
}